// DifferenceOfGaussians_33019708571908
// MI455X (gfx1250) — compile-verified
//
#include <hip/hip_runtime.h>
#include <hip/hip_bf16.h>

// CDNA5 / gfx1250: wave32, WMMA f32 16x16x4.
typedef float v2f __attribute__((ext_vector_type(2)));
typedef float v8f __attribute__((ext_vector_type(8)));

#define IMH 1024
#define IMW 1024
#define NSIG 51        // sigma bins + 1
#define KLEN 101       // 1D kernel length
#define RAD  50
#define KPAD 104       // taps padded to multiple of 4 (WMMA K=4 chunks)
#define SPAD 64        // sigma channels padded to 4 WMMA M-tiles
#define NDOG 50
#define HWSZ ((size_t)IMH * IMW)
#define P1H  8                     // h rows per block in pass 1
#define P1PR (P1H + KPAD - 1)      // 111 patch rows

static __device__ __forceinline__ v8f wmma4(v2f a, v2f b, v8f c) {
  return __builtin_amdgcn_wmma_f32_16x16x4_f32(
      /*neg_a=*/false, a, /*neg_b=*/false, b,
      /*c_mod=*/(short)0, c, /*reuse_a=*/false, /*reuse_b=*/false);
}

// ---------------------------------------------------------------------------
// Pass 1: vertical separable conv for all 51 sigmas via WMMA.
// D[16 sig x 16 w] += A[16 sig x 4 taps] x B[4 taps x 16 w]
// Block = 8 waves = 8 consecutive h rows x one 16-wide w group.
// Image patch (111 rows x 16 cols) staged in LDS -> branch-free inner loop.
// ---------------------------------------------------------------------------
__global__ __launch_bounds__(256) void dog_vert_wmma(
    const float* __restrict__ img, const float* __restrict__ k1,
    float* __restrict__ y) {
  __shared__ float lA[SPAD * KPAD];      // padded kernel matrix, 26.6 KB
  __shared__ float patch[P1PR * 16];     // image patch, 7.1 KB
  const int tid = threadIdx.x;
  const int wg = blockIdx.x & 63;        // w group (64 per row)
  const int hg = blockIdx.x >> 6;        // h group (128 of 8 rows)
  const int h0 = hg * P1H;
  const int w0 = wg << 4;

  for (int i = tid; i < SPAD * KPAD; i += 256) {
    int s = i / KPAD, t = i % KPAD;
    lA[i] = (s < NSIG && t < KLEN) ? k1[s * KLEN + t] : 0.0f;
  }
  for (int i = tid; i < P1PR * 16; i += 256) {
    const int r = i >> 4, c = i & 15;
    const int row = h0 - RAD + r;
    patch[i] = (row >= 0 && row < IMH) ? img[(size_t)row * IMW + w0 + c] : 0.0f;
  }
  __syncthreads();

  const int wave = tid >> 5, lane = tid & 31;
  const int h = h0 + wave;
  const int n = lane & 15;               // N (w) / M (sigma) sub-index
  const int kh = (lane >> 4) << 1;       // K base per lane half: 0 or 2

  v8f acc[4] = {v8f{}, v8f{}, v8f{}, v8f{}};
  for (int it = 0; it < KPAD / 4; ++it) {
    const int t = it * 4;
    // B operand: patch rows (wave + t + kh, +1), col n  -- branch free
    const float* bp = &patch[(wave + t + kh) * 16 + n];
    v2f b;
    b.x = bp[0];
    b.y = bp[16];
    // A operand per sigma tile: 16 sigma rows x 4 taps from LDS
#pragma unroll
    for (int st = 0; st < 4; ++st) {
      const float* ap = &lA[(st * 16 + n) * KPAD + t + kh];
      v2f a;
      a.x = ap[0];
      a.y = ap[1];
      acc[st] = wmma4(a, b, acc[st]);
    }
  }

  // Store C/D layout: VGPR j -> sigma row st*16 + j (lanes 0-15) / +8 (16-31).
  // Tiles 0..2 are always < 51; tile 3 is valid only for lanes 0-15, j < 3.
  const int half = lane >> 4;
  const size_t base = (size_t)h * IMW + w0 + n;
#pragma unroll
  for (int st = 0; st < 3; ++st)
#pragma unroll
    for (int j = 0; j < 8; ++j)
      y[(size_t)(st * 16 + j + 8 * half) * HWSZ + base] = acc[st][j];
  if (half == 0) {
#pragma unroll
    for (int j = 0; j < 3; ++j)
      y[(size_t)(48 + j) * HWSZ + base] = acc[3][j];
  }
}

// ---------------------------------------------------------------------------
// Pass 2: horizontal depthwise conv as banded WMMA + fused DoG.
// z[16h x 16w] = Y[16h x 116] x Band[116 x 16], Band[p][n] = k1[s][n+100-p].
// Band kernel stored REVERSED in LDS so the two K taps are consecutive
// ascending reads. Two channels (s, s+1) per wave -> dog = (zA - zB)*sigma.
// ---------------------------------------------------------------------------
__global__ __launch_bounds__(256) void dog_horz_wmma(
    const float* __restrict__ y, const float* __restrict__ k1,
    const float* __restrict__ sigmas, float* __restrict__ dog) {
  __shared__ float krev[2][132];         // krev[ch][j] = kext[131-j]
  __shared__ float ystrip[2][16][228];   // 128 cols + 100 halo, 2 ch, 29.2 KB
  const int tid = threadIdx.x;
  const int b = blockIdx.x;
  const int wt8 = b & 7;                 // which 128-col slab
  const int ht  = (b >> 3) & 63;         // 16-row tile
  const int s   = b >> 9;                // dog channel 0..49

  for (int i = tid; i < 2 * 132; i += 256) {
    const int ch = i / 132, j = i % 132;
    const int kidx = 115 - j;            // reversed: kext index (131-j) - 16
    krev[ch][j] = (kidx >= 0 && kidx < KLEN) ? k1[(s + ch) * KLEN + kidx] : 0.0f;
  }
  const int h0 = ht << 4;
  const int wbase = wt8 << 7;
  for (int i = tid; i < 2 * 16 * 228; i += 256) {
    const int ch = i / (16 * 228);
    const int rem = i % (16 * 228);
    const int row = rem / 228, c = rem % 228;
    const int col = wbase - RAD + c;
    ystrip[ch][row][c] =
        (col >= 0 && col < IMW)
            ? y[(size_t)(s + ch) * HWSZ + (size_t)(h0 + row) * IMW + col]
            : 0.0f;
  }
  __syncthreads();

  const int wave = tid >> 5, lane = tid & 31;
  const int m = lane & 15;               // A: h row; D: w col
  const int kh = (lane >> 4) << 1;
  const int wloc = wave << 4;            // wave's 16-col tile inside slab

  v8f accA = {}, accB = {};
  for (int it = 0; it < 29; ++it) {      // 116 band columns / K=4
    const int p = it * 4 + kh;
    v2f aA, aB, bA, bB;
    const float* yp0 = &ystrip[0][m][wloc + p];
    const float* yp1 = &ystrip[1][m][wloc + p];
    aA.x = yp0[0]; aA.y = yp0[1];
    aB.x = yp1[0]; aB.y = yp1[1];
    // Band[p][n] = kext[n + 116 - p] = krev[p + 15 - n]; taps ascend in p.
    const int ri = p + 15 - m;           // in [0, 130]
    bA.x = krev[0][ri]; bA.y = krev[0][ri + 1];
    bB.x = krev[1][ri]; bB.y = krev[1][ri + 1];
    accA = wmma4(aA, bA, accA);
    accB = wmma4(aB, bB, accB);
  }

  const float sg = sigmas[s];
  const int half = lane >> 4;
  const int wcol = wbase + wloc + m;
#pragma unroll
  for (int j = 0; j < 8; ++j) {
    const int hh = h0 + j + 8 * half;
    dog[(size_t)s * HWSZ + (size_t)hh * IMW + wcol] = (accA[j] - accB[j]) * sg;
  }
}

// ---------------------------------------------------------------------------
// Pass 3: 3x3x3 max-pool (-inf padded) NMS mask, LDS tiled (3 x 18 x 66).
// ---------------------------------------------------------------------------
#define TH3 16
#define TW3 64
__global__ __launch_bounds__(256) void dog_nms(
    const float* __restrict__ dog, float* __restrict__ mask) {
  __shared__ float tile[3][TH3 + 2][TW3 + 2];
  const float NEG = -3.402823466e38f;
  const int b = blockIdx.x;
  const int wt = b & 15;                 // 1024/64 col tiles
  const int ht = (b >> 4) & 63;          // 1024/16 row tiles
  const int s  = b >> 10;                // 0..49
  const int h0 = ht * TH3, w0 = wt * TW3;
  const int tid = threadIdx.x;

  for (int i = tid; i < 3 * (TH3 + 2) * (TW3 + 2); i += 256) {
    const int ss = i / ((TH3 + 2) * (TW3 + 2));
    const int rem = i % ((TH3 + 2) * (TW3 + 2));
    const int r = rem / (TW3 + 2), c = rem % (TW3 + 2);
    const int sg = s + ss - 1;
    const int hh = h0 + r - 1, ww = w0 + c - 1;
    float v = NEG;
    if (sg >= 0 && sg < NDOG && hh >= 0 && hh < IMH && ww >= 0 && ww < IMW)
      v = dog[(size_t)sg * HWSZ + (size_t)hh * IMW + ww];
    tile[ss][r][c] = v;
  }
  __syncthreads();

  for (int i = tid; i < TH3 * TW3; i += 256) {
    const int r = i / TW3, c = i % TW3;
    const float cen = tile[1][r + 1][c + 1];
    float mx = NEG;
#pragma unroll
    for (int ds = 0; ds < 3; ++ds)
#pragma unroll
      for (int dr = 0; dr < 3; ++dr)
#pragma unroll
        for (int dc = 0; dc < 3; ++dc)
          mx = fmaxf(mx, tile[ds][r + dr][c + dc]);
    const float out = (cen == mx && cen > 0.001f) ? 1.0f : 0.0f;
    mask[(size_t)s * HWSZ + (size_t)(h0 + r) * IMW + (w0 + c)] = out;
  }
}

// ---------------------------------------------------------------------------
// d_in: [0]=image [1,1,1024,1024] f32, [1]=kernels_1d [51,101] f32,
//       [2]=sigmas [51] f32.
// d_out: dog [50,1024,1024] f32 ‖ mask [50,1024,1024] f32 (bool as 0/1).
// d_ws: vertical-pass intermediate y [51,1024,1024] f32 (~204 MB).
// ---------------------------------------------------------------------------
extern "C" void kernel_launch(void* const* d_in, const int* in_sizes, int n_in,
                              void* d_out, int out_size, void* d_ws,
                              size_t ws_size, hipStream_t stream) {
  (void)in_sizes; (void)n_in; (void)out_size; (void)ws_size;
  const float* img = (const float*)d_in[0];
  const float* k1  = (const float*)d_in[1];
  const float* sg  = (const float*)d_in[2];
  float* out  = (float*)d_out;
  float* dog  = out;                         // [50][H][W]
  float* mask = out + (size_t)NDOG * HWSZ;   // [50][H][W]
  float* y    = (float*)d_ws;                // [51][H][W]

  // Pass 1: 128 h-groups * 64 w-groups.
  dog_vert_wmma<<<(IMH / P1H) * 64, 256, 0, stream>>>(img, k1, y);
  // Pass 2: 50 channels * 64 h-tiles * 8 slabs.
  dog_horz_wmma<<<NDOG * 64 * 8, 256, 0, stream>>>(y, k1, sg, dog);
  // Pass 3: 50 * 64 * 16 tiles.
  dog_nms<<<NDOG * 64 * 16, 256, 0, stream>>>(dog, mask);
}